// SmallObjectNMS_16423954940591
// MI455X (gfx1250) — compile-verified
//
#include <hip/hip_runtime.h>
#include <hip/hip_bf16.h>
#include <stdint.h>

#define NBOX 8000
#define MOUT 1000
#define IOU_THR_C 0.3f
#define SCORE_THR_C 0.1f
#define NWORDS 125   // ceil(8000/64)

typedef __attribute__((ext_vector_type(2))) float v2f;
typedef __attribute__((ext_vector_type(8))) float v8f;

// ---- CDNA5 async global->LDS staging (ASYNCcnt path), with safe fallback ----
#if defined(__has_builtin)
#if __has_builtin(__builtin_amdgcn_global_load_async_to_lds_b128)
#define HAS_ASYNC_LDS 1
#endif
#endif

// The builtin's parameter types (from the compiler diagnostic):
//   param0: addrspace(1) pointer to int __attribute__((vector_size(16)))
//   param1: addrspace(3) pointer to same
typedef int v4i __attribute__((vector_size(16)));
typedef __attribute__((address_space(1))) v4i* gbl_v4i_p;
typedef __attribute__((address_space(3))) v4i* lds_v4i_p;

// Generic LDS pointer = {SHARED_BASE aperture (hi32), LDS byte offset (lo32)};
// the as(3) pointer is just the lo32 offset.
__device__ __forceinline__ lds_v4i_p to_lds4(const void* p) {
    return (lds_v4i_p)(unsigned int)(unsigned long long)p;
}
__device__ __forceinline__ gbl_v4i_p to_gbl4(const void* p) {
    return (gbl_v4i_p)(unsigned long long)p;
}
__device__ __forceinline__ void wait_async0() {
#if defined(__has_builtin)
#if __has_builtin(__builtin_amdgcn_s_wait_asynccnt)
    __builtin_amdgcn_s_wait_asynccnt(0);
#else
    asm volatile("s_wait_asynccnt 0" ::: "memory");
#endif
#else
    asm volatile("s_wait_asynccnt 0" ::: "memory");
#endif
}

// ---------------- K0: zero the output (poisoned by harness) ----------------
__global__ void nms_zero_kernel(float* __restrict__ out, int n) {
    int i = blockIdx.x * blockDim.x + threadIdx.x;
    if (i < n) out[i] = 0.0f;
}

// ---------------- K1: max over all box coordinates, +1 ----------------
__global__ __launch_bounds__(1024) void nms_max_kernel(const float* __restrict__ boxes,
                                                       float* __restrict__ maxOut) {
    __shared__ float red[1024];
    int tid = threadIdx.x;
    float m = 0.0f;
    for (int i = tid; i < NBOX * 4; i += 1024) m = fmaxf(m, boxes[i]);
    red[tid] = m;
    __syncthreads();
    for (int s = 512; s > 0; s >>= 1) {
        if (tid < s) red[tid] = fmaxf(red[tid], red[tid + s]);
        __syncthreads();
    }
    if (tid == 0) maxOut[0] = red[0] + 1.0f;
}

// ---------------- K2: stable descending sort by score + prep sorted arrays ----------------
// key = (~ordered(score) << 32) | index; ascending sort => descending score, ties by index.
__global__ __launch_bounds__(1024) void nms_sort_prep_kernel(
        const float* __restrict__ boxes, const float* __restrict__ scores,
        const int* __restrict__ labels, const float* __restrict__ maxcPtr,
        int* __restrict__ order, int* __restrict__ svalid,
        float* __restrict__ area, float* __restrict__ bs) {
    __shared__ unsigned long long keys[8192];   // 64 KB of the 320 KB/WGP LDS
    const int tid = threadIdx.x;
    for (int i = tid; i < 8192; i += 1024) {
        unsigned long long key;
        if (i < NBOX) {
            float sc = scores[i];
            float sk = (sc > SCORE_THR_C) ? sc : -INFINITY;
            unsigned int u = __float_as_uint(sk);
            u = (u & 0x80000000u) ? ~u : (u ^ 0x80000000u);   // ascending-ordered bits
            key = ((unsigned long long)(~u) << 32) | (unsigned int)i;
        } else {
            key = ~0ULL;                                       // pad sorts to end
        }
        keys[i] = key;
    }
    // bitonic sort of 8192 elements
    for (unsigned k = 2; k <= 8192u; k <<= 1) {
        for (unsigned j = k >> 1; j > 0; j >>= 1) {
            __syncthreads();
            for (unsigned idx = tid; idx < 8192u; idx += 1024u) {
                unsigned l = idx ^ j;
                if (l > idx) {
                    unsigned long long A = keys[idx], B = keys[l];
                    bool asc = (idx & k) == 0;
                    if ((A > B) == asc) { keys[idx] = B; keys[l] = A; }
                }
            }
        }
    }
    __syncthreads();
    const float maxc = maxcPtr[0];
    for (int i = tid; i < NBOX; i += 1024) {
        int oi = (int)(keys[i] & 0xFFFFFFFFu);
        order[i] = oi;
        float sc = scores[oi];
        svalid[i] = (sc > SCORE_THR_C) ? 1 : 0;
        float off = (float)labels[oi] * maxc;
        float x1 = boxes[oi * 4 + 0], y1 = boxes[oi * 4 + 1];
        float x2 = boxes[oi * 4 + 2], y2 = boxes[oi * 4 + 3];
        bs[i * 4 + 0] = x1 + off; bs[i * 4 + 1] = y1 + off;
        bs[i * 4 + 2] = x2 + off; bs[i * 4 + 3] = y2 + off;
        area[i] = (x2 - x1) * (y2 - y1);   // offsets cancel in differences
    }
}

// ---------------- K3: suppression bitmask via WMMA-assisted IoU ----------------
// block = 256 threads = 8 waves, covering 16 rows x 512 cols (8 x 64-bit words).
// Column boxes/areas + row boxes are staged into LDS ONCE per block via
// global_load_async_to_lds_b128 (ASYNCcnt), then reused by all 8 waves x 16 rows.
// union base term S[m][n] = area_row[m] + area_col[n] via V_WMMA_F32_16X16X4_F32
// (A = 16x4 [area,1,0,0], B = 4x16 [1;area;0;0]) -- exact f32 matrix op per tile.
__global__ __launch_bounds__(256) void nms_mask_kernel(
        const float* __restrict__ bs, const float* __restrict__ area,
        unsigned long long* __restrict__ mask) {
    __shared__ __align__(16) float rowB[16 * 4];    //  256 B: 16 row boxes
    __shared__ __align__(16) float colB[512 * 4];   // 8 KB : 512 col boxes
    __shared__ __align__(16) float colA[512];       // 2 KB : 512 col areas
    const int rowTile  = blockIdx.x >> 4;           // 500 row tiles
    const int wg       = blockIdx.x & 15;           // 16 column groups of 512
    const int rowBase  = rowTile * 16;
    const int colBlock = wg * 512;
    const int tid = threadIdx.x;

#ifdef HAS_ASYNC_LDS
    // 512 col boxes: one b128 per float4, 2 per thread
    for (int i = tid; i < 512; i += 256) {
        int c  = colBlock + i;
        int cc = (c < NBOX) ? c : (NBOX - 1);
        __builtin_amdgcn_global_load_async_to_lds_b128(
            to_gbl4(bs + (size_t)cc * 4), to_lds4(&colB[i * 4]), 0, 0);
    }
    // 512 col areas: 128 b128 transfers (4 floats each)
    if (tid < 128) {
        int c4  = colBlock + tid * 4;
        int cc4 = (c4 <= NBOX - 4) ? c4 : (NBOX - 4);
        __builtin_amdgcn_global_load_async_to_lds_b128(
            to_gbl4(area + cc4), to_lds4(&colA[tid * 4]), 0, 0);
    }
    // 16 row boxes: 16 b128 transfers
    if (tid < 16) {
        __builtin_amdgcn_global_load_async_to_lds_b128(
            to_gbl4(bs + (size_t)(rowBase + tid) * 4), to_lds4(&rowB[tid * 4]), 0, 0);
    }
    wait_async0();
#else
    for (int i = tid; i < 512; i += 256) {
        int c  = colBlock + i;
        int cc = (c < NBOX) ? c : (NBOX - 1);
        colB[i * 4 + 0] = bs[cc * 4 + 0];
        colB[i * 4 + 1] = bs[cc * 4 + 1];
        colB[i * 4 + 2] = bs[cc * 4 + 2];
        colB[i * 4 + 3] = bs[cc * 4 + 3];
        colA[i] = area[cc];
    }
    if (tid < 64) rowB[tid] = bs[rowBase * 4 + tid];
#endif
    __syncthreads();

    const int lane = tid & 31;
    const int wave = tid >> 5;
    const int wordIdx = wg * 8 + wave;
    if (wordIdx >= NWORDS) return;           // whole-wave exit, EXEC stays all-ones for WMMA
    const int n = lane & 15;

    // A fragment (16x4 f32): lanes 0-15 hold K=0,1 ; lanes 16-31 hold K=2,3 (zeros)
    v2f a;
    a.x = (lane < 16) ? area[rowBase + lane] : 0.0f;   // K=0: area of row m=lane
    a.y = (lane < 16) ? 1.0f : 0.0f;                   // K=1: ones

    // prefetch next block's row-box tile (global_prefetch_b8)
    {
        int nr = rowBase + 16 < NBOX ? rowBase + 16 : rowBase;
        __builtin_prefetch(&bs[(size_t)nr * 4], 0, 1);
    }

    unsigned long long word = 0ULL;
    for (int t = 0; t < 4; ++t) {
        int localC = wave * 64 + t * 16 + n;
        int cIdx   = colBlock + localC;
        bool inRange = (cIdx < NBOX);

        float cx1 = colB[localC * 4 + 0], cy1 = colB[localC * 4 + 1];
        float cx2 = colB[localC * 4 + 2], cy2 = colB[localC * 4 + 3];

        // B fragment (4x16 f32): lanes 0-15 hold K=0,1 ; lanes 16-31 hold K=2,3 (zeros)
        v2f b;
        b.x = (lane < 16) ? 1.0f : 0.0f;                           // K=0: ones
        b.y = (lane < 16 && inRange) ? colA[localC] : 0.0f;        // K=1: area of col n
        v8f c = {};
        c = __builtin_amdgcn_wmma_f32_16x16x4_f32(
                /*neg_a=*/false, a, /*neg_b=*/false, b,
                /*c_mod=*/(short)0, c, /*reuse_a=*/false, /*reuse_b=*/false);

        #pragma unroll
        for (int r = 0; r < 8; ++r) {
            // D layout: lanes 0-15 -> M=r ; lanes 16-31 -> M=r+8 ; N = lane&15
            int m = r + ((lane & 16) ? 8 : 0);
            float rx1 = rowB[m * 4 + 0], ry1 = rowB[m * 4 + 1];
            float rx2 = rowB[m * 4 + 2], ry2 = rowB[m * 4 + 3];
            float iw = fminf(rx2, cx2) - fmaxf(rx1, cx1);
            float ih = fminf(ry2, cy2) - fmaxf(ry1, cy1);
            iw = fmaxf(iw, 0.0f); ih = fmaxf(ih, 0.0f);
            float inter = iw * ih;
            float uni = fmaxf(c[r] - inter, 1e-9f);   // area_i + area_j - inter
            bool bit = inRange && ((rowBase + m) != cIdx) && (inter > IOU_THR_C * uni);
            unsigned int bal = __builtin_amdgcn_ballot_w32(bit);
            if (lane == r)     word |= (unsigned long long)(bal & 0xFFFFu) << (t * 16);
            if (lane == r + 8) word |= (unsigned long long)(bal >> 16)     << (t * 16);
        }
    }
    if (lane < 16) mask[(size_t)(rowBase + lane) * NWORDS + wordIdx] = word;
}

// ---------------- K4: serial greedy scan over bitmask + write outputs ----------------
__global__ __launch_bounds__(128) void nms_select_kernel(
        const float* __restrict__ boxes, const float* __restrict__ scores,
        const int* __restrict__ labels, const int* __restrict__ order,
        const int* __restrict__ svalid, const unsigned long long* __restrict__ mask,
        float* __restrict__ out) {
    __shared__ unsigned long long remv[NWORDS];
    __shared__ int s_keep;
    const int tid = threadIdx.x;
    if (tid < NWORDS) remv[tid] = 0ULL;
    int kept = 0;   // only meaningful on thread 0
    for (int i = 0; i < NBOX; ++i) {
        __syncthreads();
        if (tid == 0) {
            bool sup = (remv[i >> 6] >> (i & 63)) & 1ULL;
            int k = (svalid[i] != 0) && !sup;
            s_keep = k;
            if (k) {
                if (kept < MOUT) {
                    int oi = order[i];
                    out[kept * 4 + 0] = boxes[oi * 4 + 0];
                    out[kept * 4 + 1] = boxes[oi * 4 + 1];
                    out[kept * 4 + 2] = boxes[oi * 4 + 2];
                    out[kept * 4 + 3] = boxes[oi * 4 + 3];
                    out[4 * MOUT + kept] = scores[oi];
                    out[5 * MOUT + kept] = (float)labels[oi];
                }
                kept++;
            }
        }
        __syncthreads();
        if (s_keep && tid < NWORDS)
            remv[tid] |= mask[(size_t)i * NWORDS + tid];
    }
}

extern "C" void kernel_launch(void* const* d_in, const int* in_sizes, int n_in,
                              void* d_out, int out_size, void* d_ws, size_t ws_size,
                              hipStream_t stream) {
    const float* boxes  = (const float*)d_in[0];
    const float* scores = (const float*)d_in[1];
    const int*   labels = (const int*)d_in[2];
    float* out = (float*)d_out;

    char* ws = (char*)d_ws;
    float* maxc  = (float*)(ws);
    int*   order = (int*)(ws + 256);
    int*   svalid= (int*)(ws + 256 + 32768);
    float* area  = (float*)(ws + 256 + 2 * 32768);
    float* bs    = (float*)(ws + 256 + 3 * 32768);
    unsigned long long* mask =
        (unsigned long long*)(ws + 256 + 3 * 32768 + 131072);   // 8 MB suppression matrix

    nms_zero_kernel<<<(6 * MOUT + 255) / 256, 256, 0, stream>>>(out, 6 * MOUT);
    nms_max_kernel<<<1, 1024, 0, stream>>>(boxes, maxc);
    nms_sort_prep_kernel<<<1, 1024, 0, stream>>>(boxes, scores, labels, maxc,
                                                 order, svalid, area, bs);
    nms_mask_kernel<<<500 * 16, 256, 0, stream>>>(bs, area, mask);
    nms_select_kernel<<<1, 128, 0, stream>>>(boxes, scores, labels, order, svalid, mask, out);
}